// EdgePooling_65300682769103
// MI455X (gfx1250) — compile-verified
//
#include <hip/hip_runtime.h>
#include <hip/hip_bf16.h>

typedef __attribute__((ext_vector_type(2))) float v2f;
typedef __attribute__((ext_vector_type(8))) float v8f;

#ifndef INF_F
#define INF_F __builtin_inff()
#endif

// ---------------------------------------------------------------------------
// Step 1: edge scoring via V_WMMA_F32_16X16X4_F32.
// One wave (32 lanes) scores 16 edges. A = 16 gathered rows x K=4 per step
// (ISA layout: VGPR0 = K0 (lanes 0-15) / K2 (lanes 16-31), VGPR1 = K1/K3).
// B[k][n] = W[k] broadcast across all 16 columns -> C[m][0] = dot(feat_m, W).
// ---------------------------------------------------------------------------
__global__ __launch_bounds__(32)
void ep_edge_score_wmma(const float* __restrict__ x, const float* __restrict__ W,
                        const int* __restrict__ src, const int* __restrict__ dst,
                        float* __restrict__ s, int E, int F) {
  const int lane = threadIdx.x;            // 0..31, wave32
  const int m    = lane & 15;              // matrix row (edge within tile)
  const int hi   = lane >> 4;              // 0: K0/K1, 1: K2/K3
  const int tile = blockIdx.x;
  int e  = tile * 16 + m;
  int ec = (e < E) ? e : 0;                // clamp; writes guarded below
  const float* xa = x + (long)src[ec] * F;
  const float* xb = x + (long)dst[ec] * F;
  const int koff = hi ? 2 : 0;

  v8f c = {};
  for (int k = 0; k < F; k += 4) {         // src half: W[0..F)
    v2f A, B;
    A.x = xa[k + koff];     A.y = xa[k + koff + 1];
    B.x = W[k + koff];      B.y = W[k + koff + 1];
    c = __builtin_amdgcn_wmma_f32_16x16x4_f32(false, A, false, B, (short)0, c, false, false);
  }
  for (int k = 0; k < F; k += 4) {         // dst half: W[F..2F)
    v2f A, B;
    A.x = xb[k + koff];     A.y = xb[k + koff + 1];
    B.x = W[F + k + koff];  B.y = W[F + k + koff + 1];
    c = __builtin_amdgcn_wmma_f32_16x16x4_f32(false, A, false, B, (short)0, c, false, false);
  }
  // C layout: VGPR r, lanes 0-15 -> M=r, lanes 16-31 -> M=r+8; N = lane&15.
  // Column N=0 lives in lanes 0 and 16.
  if (m == 0) {
    const float inv = 1.0f / __builtin_sqrtf(2.0f * (float)F);  // e3nn path norm
    int base = tile * 16 + hi * 8;
#pragma unroll
    for (int r = 0; r < 8; ++r) {
      int ee = base + r;
      if (ee < E) s[ee] = c[r] * inv;
    }
  }
}

// ---------------------------------------------------------------------------
// Init / utility kernels
// ---------------------------------------------------------------------------
__global__ void ep_zero_f(float* __restrict__ p, int n) {
  int i = blockIdx.x * blockDim.x + threadIdx.x;
  if (i < n) p[i] = 0.0f;
}

__global__ void ep_init_nodes(unsigned* __restrict__ smax, float* __restrict__ denom,
                              int* __restrict__ mask, int* __restrict__ cluster,
                              float* __restrict__ cscore, float* __restrict__ counts,
                              int N) {
  int i = blockIdx.x * blockDim.x + threadIdx.x;
  if (i >= N) return;
  smax[i]   = 0u;        // ordered-uint -inf sentinel (below any finite transform)
  denom[i]  = 0.0f;
  mask[i]   = 1;
  cluster[i]= 0;
  cscore[i] = 1.0f;      // reference: cscore defaults to 1, selected overwritten
  counts[i] = 0.0f;
}

// monotonic float->uint transform for atomic max
__device__ __forceinline__ unsigned ep_f2u(float f) {
  unsigned u = __float_as_uint(f);
  return (u & 0x80000000u) ? ~u : (u | 0x80000000u);
}
__device__ __forceinline__ float ep_u2f(unsigned t) {
  return (t & 0x80000000u) ? __uint_as_float(t & 0x7FFFFFFFu) : __uint_as_float(~t);
}

__global__ void ep_segmax(const float* __restrict__ s, const int* __restrict__ dst,
                          unsigned* __restrict__ smax, int E) {
  int e = blockIdx.x * blockDim.x + threadIdx.x;
  if (e >= E) return;
  atomicMax(&smax[dst[e]], ep_f2u(s[e]));
}

__global__ void ep_expsum(const float* __restrict__ s, const int* __restrict__ dst,
                          const unsigned* __restrict__ smax, float* __restrict__ ex,
                          float* __restrict__ denom, int E) {
  int e = blockIdx.x * blockDim.x + threadIdx.x;
  if (e >= E) return;
  int d = dst[e];
  float v = __expf(s[e] - ep_u2f(smax[d]));
  ex[e] = v;
  atomicAdd(&denom[d], v);
}

__global__ void ep_score_fill(const float* __restrict__ ex, const float* __restrict__ denom,
                              const int* __restrict__ dst, float* __restrict__ score,
                              float* __restrict__ skey, int* __restrict__ sval,
                              int E, int NP2) {
  int i = blockIdx.x * blockDim.x + threadIdx.x;
  if (i >= NP2) return;
  if (i < E) {
    float sc = ex[i] / (denom[dst[i]] + 1e-16f);
    score[i] = sc;
    skey[i]  = sc;
    sval[i]  = i;
  } else {
    skey[i] = -INF_F;          // pads sort last (descending key order)
    sval[i] = 0x7FFFFFFF;
  }
}

// ---------------------------------------------------------------------------
// Bitonic sort passes. Comparator reproduces stable argsort:
//   float version: descending key, ascending index on ties  (argsort(-score))
//   int   version: ascending key,  ascending index on ties  (argsort(keys))
// ---------------------------------------------------------------------------
__global__ void ep_bitonic_f(float* __restrict__ key, int* __restrict__ val,
                             int k, int j, int n) {
  int i = blockIdx.x * blockDim.x + threadIdx.x;
  if (i >= n) return;
  int p = i ^ j;
  if (p <= i) return;
  float ka = key[i], kb = key[p];
  int   va = val[i], vb = val[p];
  bool aFirst = (ka > kb) || (ka == kb && va < vb);   // desired order at [i]
  bool up = ((i & k) == 0);
  if (up ? !aFirst : aFirst) {
    key[i] = kb; key[p] = ka;
    val[i] = vb; val[p] = va;
  }
}

__global__ void ep_bitonic_i(int* __restrict__ key, int* __restrict__ val,
                             int k, int j, int n) {
  int i = blockIdx.x * blockDim.x + threadIdx.x;
  if (i >= n) return;
  int p = i ^ j;
  if (p <= i) return;
  int ka = key[i], kb = key[p];
  int va = val[i], vb = val[p];
  bool aFirst = (ka < kb) || (ka == kb && va < vb);
  bool up = ((i & k) == 0);
  if (up ? !aFirst : aFirst) {
    key[i] = kb; key[p] = ka;
    val[i] = vb; val[p] = va;
  }
}

// ---------------------------------------------------------------------------
// Sequential greedy matching (inherently serial per the reference scan).
// ---------------------------------------------------------------------------
__global__ void ep_greedy(const int* __restrict__ src, const int* __restrict__ dst,
                          const int* __restrict__ sval, const float* __restrict__ score,
                          int* __restrict__ mask, int* __restrict__ cluster,
                          float* __restrict__ cscore, float* __restrict__ out_ncl,
                          int N, int E, int NP2) {
  if (threadIdx.x != 0 || blockIdx.x != 0) return;
  int nsel = 0;
  for (int k = 0; k < NP2; ++k) {
    int e = sval[k];
    if (e < 0 || e >= E) continue;       // padding
    int a = src[e], b = dst[e];
    if (mask[a] & mask[b]) {
      mask[a] = 0; mask[b] = 0;          // a==b self-loop handled identically
      cluster[a] = nsel; cluster[b] = nsel;
      cscore[nsel] = score[e];
      ++nsel;
    }
  }
  int j = 0;
  for (int n = 0; n < N; ++n)
    if (mask[n]) { cluster[n] = nsel + j; ++j; }   // cscore stays 1.0
  *out_ncl = (float)(nsel + j);
}

// ---------------------------------------------------------------------------
// Cluster scatter-reduces + finalization
// ---------------------------------------------------------------------------
__global__ void ep_scatter_x(const float* __restrict__ x, const int* __restrict__ cluster,
                             float* __restrict__ sumx, int N, int F) {
  int i = blockIdx.x * blockDim.x + threadIdx.x;
  if (i >= N * F) return;
  int n = i / F, f = i - n * F;
  atomicAdd(&sumx[(long)cluster[n] * F + f], x[i]);
}

__global__ void ep_scatter_pos(const float* __restrict__ pos, const int* __restrict__ cluster,
                               float* __restrict__ sumpos, float* __restrict__ counts, int N) {
  int n = blockIdx.x * blockDim.x + threadIdx.x;
  if (n >= N) return;
  int c = cluster[n];
  atomicAdd(&sumpos[c * 3 + 0], pos[n * 3 + 0]);
  atomicAdd(&sumpos[c * 3 + 1], pos[n * 3 + 1]);
  atomicAdd(&sumpos[c * 3 + 2], pos[n * 3 + 2]);
  atomicAdd(&counts[c], 1.0f);
}

__global__ void ep_final_x(const float* __restrict__ sumx, const float* __restrict__ cscore,
                           float* __restrict__ out_x, int N, int F) {
  int i = blockIdx.x * blockDim.x + threadIdx.x;
  if (i >= N * F) return;
  out_x[i] = sumx[i] * cscore[i / F];
}

__global__ void ep_final_pos(const float* __restrict__ sumpos, const float* __restrict__ counts,
                             float* __restrict__ out_pos, int N) {
  int i = blockIdx.x * blockDim.x + threadIdx.x;
  if (i >= N * 3) return;
  float c = counts[i / 3];
  out_pos[i] = sumpos[i] / fmaxf(c, 1.0f);
}

// ---------------------------------------------------------------------------
// Edge coalescing
// ---------------------------------------------------------------------------
__global__ void ep_coalesce_keys(const int* __restrict__ src, const int* __restrict__ dst,
                                 const int* __restrict__ cluster, int* __restrict__ ikey,
                                 int* __restrict__ ival, int N, int E, int NP2) {
  int i = blockIdx.x * blockDim.x + threadIdx.x;
  if (i >= NP2) return;
  if (i < E) {
    int ci = cluster[src[i]], cj = cluster[dst[i]];
    ikey[i] = ci * N + cj;                 // N=10000 -> fits int32
    ival[i] = i;
  } else {
    ikey[i] = 0x7FFFFFFF;
    ival[i] = 0x7FFFFFFF;
  }
}

__global__ void ep_edge_final(const int* __restrict__ src, const int* __restrict__ dst,
                              const int* __restrict__ cluster, const int* __restrict__ ikey,
                              const int* __restrict__ ival, const float* __restrict__ out_pos,
                              float* __restrict__ out_ei, float* __restrict__ out_uniq,
                              float* __restrict__ out_evec, int E) {
  int t = blockIdx.x * blockDim.x + threadIdx.x;
  if (t >= E) return;
  int e  = ival[t];
  int ci = cluster[src[e]], cj = cluster[dst[e]];
  out_ei[t]     = (float)ci;
  out_ei[E + t] = (float)cj;
  float u = (t == 0 || ikey[t] != ikey[t - 1]) ? 1.0f : 0.0f;
  out_uniq[t] = u;
  out_evec[t * 3 + 0] = (out_pos[ci * 3 + 0] - out_pos[cj * 3 + 0]) * u;
  out_evec[t * 3 + 1] = (out_pos[ci * 3 + 1] - out_pos[cj * 3 + 1]) * u;
  out_evec[t * 3 + 2] = (out_pos[ci * 3 + 2] - out_pos[cj * 3 + 2]) * u;
}

__global__ void ep_batch_scatter(const int* __restrict__ batch, const int* __restrict__ cluster,
                                 float* __restrict__ out_batch, int N) {
  int n = blockIdx.x * blockDim.x + threadIdx.x;
  if (n >= N) return;
  out_batch[cluster[n]] = (float)batch[n];
}

// ---------------------------------------------------------------------------
// Launcher
// ---------------------------------------------------------------------------
extern "C" void kernel_launch(void* const* d_in, const int* in_sizes, int n_in,
                              void* d_out, int out_size, void* d_ws, size_t ws_size,
                              hipStream_t stream) {
  const float* x     = (const float*)d_in[0];
  const float* pos   = (const float*)d_in[1];
  const float* W     = (const float*)d_in[2];
  const int*   ei    = (const int*)d_in[3];
  const int*   batch = (const int*)d_in[4];

  const int N = in_sizes[4];
  const int F = in_sizes[0] / N;
  const int E = in_sizes[3] / 2;
  const int* src = ei;
  const int* dst = ei + E;

  int NP2 = 1;
  while (NP2 < E) NP2 <<= 1;

  // ---- workspace carve (4-byte types, 256B-aligned chunks) ----
  char* wp = (char*)d_ws;
  auto carve = [&](size_t bytes) { char* p = wp; wp += (bytes + 255) & ~(size_t)255; return p; };
  float*    s      = (float*)   carve((size_t)E * 4);
  unsigned* smax   = (unsigned*)carve((size_t)N * 4);
  float*    denom  = (float*)   carve((size_t)N * 4);
  float*    ex     = (float*)   carve((size_t)E * 4);
  float*    score  = (float*)   carve((size_t)E * 4);
  float*    skey   = (float*)   carve((size_t)NP2 * 4);
  int*      sval   = (int*)     carve((size_t)NP2 * 4);
  int*      mask   = (int*)     carve((size_t)N * 4);
  int*      clus   = (int*)     carve((size_t)N * 4);
  float*    cscore = (float*)   carve((size_t)N * 4);
  float*    counts = (float*)   carve((size_t)N * 4);
  float*    sumx   = (float*)   carve((size_t)N * F * 4);
  float*    sumpos = (float*)   carve((size_t)N * 3 * 4);
  int*      ikey   = (int*)     carve((size_t)NP2 * 4);
  int*      ival   = (int*)     carve((size_t)NP2 * 4);

  // ---- output layout (reference tuple order, flattened) ----
  float* out       = (float*)d_out;
  float* out_x     = out;                          // N*F
  float* out_pos   = out_x   + (size_t)N * F;      // N*3
  float* out_ei    = out_pos + (size_t)N * 3;      // 2*E
  float* out_uniq  = out_ei  + (size_t)2 * E;      // E
  float* out_evec  = out_uniq + (size_t)E;         // E*3
  float* out_batch = out_evec + (size_t)E * 3;     // N
  float* out_ncl   = out_batch + (size_t)N;        // 1

  const int B = 256;
  auto g = [&](long n) { return dim3((unsigned)((n + B - 1) / B)); };

  // init
  ep_init_nodes<<<g(N), B, 0, stream>>>(smax, denom, mask, clus, cscore, counts, N);
  ep_zero_f<<<g((long)N * F), B, 0, stream>>>(sumx, N * F);
  ep_zero_f<<<g((long)N * 3), B, 0, stream>>>(sumpos, N * 3);

  // 1) edge scoring on the WMMA pipe (one wave per 16-edge tile)
  ep_edge_score_wmma<<<dim3((E + 15) / 16), dim3(32), 0, stream>>>(x, W, src, dst, s, E, F);

  // 2) segment softmax over dst
  ep_segmax<<<g(E), B, 0, stream>>>(s, dst, smax, E);
  ep_expsum<<<g(E), B, 0, stream>>>(s, dst, smax, ex, denom, E);
  ep_score_fill<<<g(NP2), B, 0, stream>>>(ex, denom, dst, score, skey, sval, E, NP2);

  // 3a) stable argsort(-score): bitonic over NP2
  for (int k = 2; k <= NP2; k <<= 1)
    for (int j = k >> 1; j > 0; j >>= 1)
      ep_bitonic_f<<<g(NP2), B, 0, stream>>>(skey, sval, k, j, NP2);

  // 3b) sequential greedy matching + cluster ids
  ep_greedy<<<dim3(1), dim3(32), 0, stream>>>(src, dst, sval, score, mask, clus, cscore,
                                              out_ncl, N, E, NP2);

  // 4) cluster scatter-reduces + finalization
  ep_scatter_x<<<g((long)N * F), B, 0, stream>>>(x, clus, sumx, N, F);
  ep_scatter_pos<<<g(N), B, 0, stream>>>(pos, clus, sumpos, counts, N);
  ep_final_x<<<g((long)N * F), B, 0, stream>>>(sumx, cscore, out_x, N, F);
  ep_final_pos<<<g((long)N * 3), B, 0, stream>>>(sumpos, counts, out_pos, N);

  // 5) coalesce edges: stable sort by ci*N+cj, uniq mask, edge vectors
  ep_coalesce_keys<<<g(NP2), B, 0, stream>>>(src, dst, clus, ikey, ival, N, E, NP2);
  for (int k = 2; k <= NP2; k <<= 1)
    for (int j = k >> 1; j > 0; j >>= 1)
      ep_bitonic_i<<<g(NP2), B, 0, stream>>>(ikey, ival, k, j, NP2);
  ep_edge_final<<<g(E), B, 0, stream>>>(src, dst, clus, ikey, ival, out_pos,
                                        out_ei, out_uniq, out_evec, E);

  // 6) new_batch
  ep_zero_f<<<g(N), B, 0, stream>>>(out_batch, N);
  ep_batch_scatter<<<g(N), B, 0, stream>>>(batch, clus, out_batch, N);
}